// ValueIterNet_55224689492553
// MI455X (gfx1250) — compile-verified
//
#include <hip/hip_runtime.h>
#include <hip/hip_bf16.h>

#define NDIM1   41
#define NPTS    2825761            // 41^4
#define NA      5
#define TILE    256
#define CHUNK   (TILE * NA)        // 1280 floats per staged chunk
#define NCHUNK  5                  // 4 dsdt dims + cost
#define TILES   ((NPTS + TILE - 1) / TILE)   // 11039
#define STAGE_ISSUES 25            // async b32 issues per thread per tile

// ---- CDNA5 async global->LDS copy (ASYNCcnt-tracked), non-temporal ---------
// NT hint: dsdt/cost are a 283 MB single-use stream (> 192 MB L2); keep the
// reused J array resident in L2 instead.
__device__ __forceinline__ void async_cp_f32_nt(unsigned lds_off, const float* g) {
    asm volatile("global_load_async_to_lds_b32 %0, %1, off th:TH_LOAD_NT"
                 :: "v"(lds_off), "v"(g)
                 : "memory");
}

// Stage one tile's dsdt (4 chunks) + cost (1 chunk) into LDS.
// Exactly STAGE_ISSUES async instructions per thread (addresses clamped so the
// partial last tile never reads out of bounds and the count stays constant).
__device__ __forceinline__ void stage_tile(const float* __restrict__ dsdt,
                                           const float* __restrict__ cost,
                                           int t, float* lbuf, int tid) {
    const int maxIdx = (NPTS - t * TILE) * NA - 1;  // last valid element of this tile's chunk
#pragma unroll
    for (int c = 0; c < NCHUNK; ++c) {
        const float* g = (c < 4)
            ? dsdt + ((size_t)c * NPTS + (size_t)t * TILE) * NA
            : cost + (size_t)t * TILE * NA;
        unsigned lds0 = (unsigned)(size_t)(&lbuf[c * CHUNK]);
#pragma unroll
        for (int k = 0; k < CHUNK / TILE; ++k) {    // 5 iterations
            int idx = k * TILE + tid;
            int gi  = idx > maxIdx ? maxIdx : idx;
            async_cp_f32_nt(lds0 + 4u * (unsigned)idx, g + gi);
        }
    }
}

// One-sided differences matching the reference concat semantics exactly.
//  up[i]   = (J[i+1]-J[i])/(2h) for 1<=i<=39 ; edges: periodic -> (J[1]-J[40])/(2h), else 0
//  down[i] = (J[i]-J[i-1])/(2h) for 1<=i<=39 ; edges: periodic -> (J[0]-J[39])/(2h), else 0
__device__ __forceinline__ void diffs(const float* __restrict__ J, int base,
                                      int c, int s, bool periodic, float twoh,
                                      float& up, float& dn) {
    if (c >= 1 && c <= NDIM1 - 2) {
        float Jc = J[base];
        up = (J[base + s] - Jc) / twoh;
        dn = (Jc - J[base - s]) / twoh;
    } else if (periodic) {
        int b0 = base - c * s;                       // coordinate 0 along this dim
        up = (J[b0 + s] - J[b0 + (NDIM1 - 1) * s]) / twoh;
        dn = (J[b0]     - J[b0 + (NDIM1 - 2) * s]) / twoh;
    } else {
        up = 0.0f; dn = 0.0f;
    }
}

__global__ __launch_bounds__(TILE)
void ValueIterNet_55224689492553_kernel(const float* __restrict__ J,
                                        const float* __restrict__ dsdt,
                                        const float* __restrict__ cost,
                                        float* __restrict__ dJdt,
                                        int*   __restrict__ aout) {
    __shared__ float lbuf[2 * NCHUNK * CHUNK];       // 51.2 KB, double buffered
    const int tid    = threadIdx.x;
    const int stride = gridDim.x;

    int t = blockIdx.x;
    if (t >= TILES) return;

    int phase = 0;
    stage_tile(dsdt, cost, t, &lbuf[0], tid);        // prologue: fill buffer 0

    for (; t < TILES; t += stride) {
        const int  tn  = t + stride;
        const bool pre = (tn < TILES);
        if (pre) {
            stage_tile(dsdt, cost, tn, &lbuf[(1 - phase) * NCHUNK * CHUNK], tid);
            // current tile's 25 copies are the oldest -> done once <=25 remain
            asm volatile("s_wait_asynccnt 25" ::: "memory");
        } else {
            asm volatile("s_wait_asynccnt 0" ::: "memory");
        }
        __syncthreads();                             // all waves' copies landed in LDS

        float* lb = &lbuf[phase * NCHUNK * CHUNK];
        int npts = NPTS - t * TILE; if (npts > TILE) npts = TILE;
        if (tid < npts) {
            const int pt = t * TILE + tid;           // flat [i0][i1][i2][i3] index
            unsigned u = (unsigned)pt;
            int i3 = (int)(u % 41u); unsigned q = u / 41u;
            int i2 = (int)(q % 41u); q /= 41u;
            int i1 = (int)(q % 41u);
            int i0 = (int)(q / 41u);

            float up0, dn0, up1, dn1, up2, dn2, up3, dn3;
            diffs(J, pt, i0, 68921, true,  0.31415926f, up0, dn0);  // 41^3 stride
            diffs(J, pt, i1, 1681,  true,  0.31415926f, up1, dn1);  // 41^2
            diffs(J, pt, i2, 41,    false, 0.8f,        up2, dn2);
            diffs(J, pt, i3, 1,     false, 0.8f,        up3, dn3);

            float best = __builtin_inff();
            int   ba   = 0;
#pragma unroll
            for (int a = 0; a < NA; ++a) {
                float g0 = lb[0 * CHUNK + tid * NA + a];
                float g1 = lb[1 * CHUNK + tid * NA + a];
                float g2 = lb[2 * CHUNK + tid * NA + a];
                float g3 = lb[3 * CHUNK + tid * NA + a];
                float s  = lb[4 * CHUNK + tid * NA + a];   // cost
                s += up0 * fmaxf(g0, 0.0f) + dn0 * fminf(g0, 0.0f);
                s += up1 * fmaxf(g1, 0.0f) + dn1 * fminf(g1, 0.0f);
                s += up2 * fmaxf(g2, 0.0f) + dn2 * fminf(g2, 0.0f);
                s += up3 * fmaxf(g3, 0.0f) + dn3 * fminf(g3, 0.0f);
                if (s < best) { best = s; ba = a; }        // first-occurrence argmin
            }
            // outputs are write-once, never re-read: NT stores keep L2 for J
            __builtin_nontemporal_store(best, &dJdt[pt]);
            __builtin_nontemporal_store(ba,   &aout[pt]);
        }
        __syncthreads();                             // reads done before buffer reuse
        phase ^= 1;
    }
}

extern "C" void kernel_launch(void* const* d_in, const int* in_sizes, int n_in,
                              void* d_out, int out_size, void* d_ws, size_t ws_size,
                              hipStream_t stream) {
    const float* J    = (const float*)d_in[0];   // [41,41,41,41]
    const float* dsdt = (const float*)d_in[1];   // [4,41,41,41,41,5]
    const float* cost = (const float*)d_in[2];   // [41,41,41,41,5]
    float* dJdt = (float*)d_out;                 // first 41^4 floats
    int*   aout = (int*)d_out + NPTS;            // argmin indices (int32 bits)

    int nb = TILES < 4096 ? TILES : 4096;        // persistent blocks, grid-stride over tiles
    ValueIterNet_55224689492553_kernel<<<dim3(nb), dim3(TILE), 0, stream>>>(
        J, dsdt, cost, dJdt, aout);
}